// ChADALINE_56762287784249
// MI455X (gfx1250) — compile-verified
//
#include <hip/hip_runtime.h>

// Choquet integral with cardinality measure == mean over last axis, so the
// reference collapses to: sigmoid( (x @ W + colsum(bias)) / IN ).
// GEMM done with V_WMMA_F32_16X16X4_F32 (full-precision f32 matrix pipe).

typedef float v2f __attribute__((ext_vector_type(2)));
typedef float v8f __attribute__((ext_vector_type(8)));

#define BATCH 256
#define IN_SZ 1024
#define OUT_SZ 1024

// ---------------------------------------------------------------------------
// Kernel 1: bsum[o] = sum_i bias[i][o]   (coalesced column reduction)
// ---------------------------------------------------------------------------
__global__ void chq_bias_colsum(const float* __restrict__ bias,
                                float* __restrict__ bsum) {
    const int o = blockIdx.x * blockDim.x + threadIdx.x;  // 0..OUT_SZ-1
    float s = 0.0f;
#pragma unroll 8
    for (int i = 0; i < IN_SZ; ++i) {
        s += bias[i * OUT_SZ + o];
    }
    bsum[o] = s;
}

// ---------------------------------------------------------------------------
// Kernel 2: C = X @ W via v_wmma_f32_16x16x4_f32, epilogue:
//           out = sigmoid((C + bsum) / IN_SZ)
//
// One wave -> 16(M) x 32(N) output strip, two f32 accumulators.
// Block = 128 threads = 4 waves. Grid = (OUT/32/4, BATCH/16) = (8,16).
//
// Fragment layouts (ISA 7.12.2, wave32):
//   A 16x4 : lane(half,lr) vgpr v -> A[lr][k + v + 2*half]      (float2 load)
//   B 4x16 : lane(half,lr) vgpr v -> B[k + v + 2*half][n + lr]
//   C 16x16: lane(half,lr) vgpr v -> C[v + 8*half][lr]
// ---------------------------------------------------------------------------
__global__ void __launch_bounds__(128)
chq_wmma_gemm(const float* __restrict__ X,     // [BATCH, IN]
              const float* __restrict__ W,     // [IN, OUT]
              const float* __restrict__ bsum,  // [OUT]
              float* __restrict__ out) {       // [BATCH, OUT]
    const int wave = threadIdx.x >> 5;   // 0..3
    const int lane = threadIdx.x & 31;
    const int half = lane >> 4;          // 0 or 1
    const int lr   = lane & 15;          // 0..15

    const int m0 = blockIdx.y * 16;                    // M tile origin
    const int n0 = (blockIdx.x * 4 + wave) * 32;       // N strip origin

    v8f acc0 = {};
    v8f acc1 = {};

    // Per-lane A base: row (m0+lr), k offset 2*half; contiguous pair -> b64 load.
    const float* __restrict__ arow = X + (m0 + lr) * IN_SZ + 2 * half;
    // Per-lane B columns.
    const float* __restrict__ bcol0 = W + n0 + lr;
    const float* __restrict__ bcol1 = W + n0 + 16 + lr;

#pragma unroll 4
    for (int k = 0; k < IN_SZ; k += 4) {
        v2f a = *(const v2f*)(arow + k);      // A[lr][k+2h], A[lr][k+2h+1]

        const int kb = k + 2 * half;
        v2f b0, b1;
        b0.x = bcol0[(kb    ) * OUT_SZ];
        b0.y = bcol0[(kb + 1) * OUT_SZ];
        b1.x = bcol1[(kb    ) * OUT_SZ];
        b1.y = bcol1[(kb + 1) * OUT_SZ];

        // (neg_a, A, neg_b, B, c_mod, C, reuse_a, reuse_b)
        acc0 = __builtin_amdgcn_wmma_f32_16x16x4_f32(
            false, a, false, b0, (short)0, acc0, false, false);
        acc1 = __builtin_amdgcn_wmma_f32_16x16x4_f32(
            false, a, false, b1, (short)0, acc1, false, false);
    }

    // Fused epilogue: add bias column-sum, divide by n, sigmoid.
    const float inv_n = 1.0f / (float)IN_SZ;
    const float bs0 = bsum[n0 + lr];
    const float bs1 = bsum[n0 + 16 + lr];

#pragma unroll
    for (int v = 0; v < 8; ++v) {
        const int row = m0 + v + 8 * half;
        const float t0 = (acc0[v] + bs0) * inv_n;
        const float t1 = (acc1[v] + bs1) * inv_n;
        out[row * OUT_SZ + n0 + lr]      = 1.0f / (1.0f + __expf(-t0));
        out[row * OUT_SZ + n0 + 16 + lr] = 1.0f / (1.0f + __expf(-t1));
    }
}

// ---------------------------------------------------------------------------
extern "C" void kernel_launch(void* const* d_in, const int* in_sizes, int n_in,
                              void* d_out, int out_size, void* d_ws, size_t ws_size,
                              hipStream_t stream) {
    const float* x    = (const float*)d_in[0];   // [256, 1, 1024]
    const float* w    = (const float*)d_in[1];   // [1024, 1024]
    const float* bias = (const float*)d_in[2];   // [1024, 1024]
    float* out  = (float*)d_out;                 // [256, 1024]
    float* bsum = (float*)d_ws;                  // 1024 floats of scratch

    (void)in_sizes; (void)n_in; (void)out_size; (void)ws_size;

    chq_bias_colsum<<<OUT_SZ / 256, 256, 0, stream>>>(bias, bsum);

    dim3 grid(OUT_SZ / 32 / 4, BATCH / 16);      // (8, 16) blocks
    chq_wmma_gemm<<<grid, 128, 0, stream>>>(x, w, bsum, out);
}